// PhysicsLoss_38010460570135
// MI455X (gfx1250) — compile-verified
//
#include <hip/hip_runtime.h>
#include <hip/hip_bf16.h>

#define N_NODES  100000
#define N_EDGES  3200000
#define EPS_V    1e-6f

typedef __attribute__((ext_vector_type(2))) float v2f;
typedef __attribute__((ext_vector_type(8))) float v8f;

// Wave32 all-reduce using V_WMMA_F32_16X16X4_F32.
// A(16x4): lane m (0-15) supplies A[m][0]=x, A[m][1]=0; lane m+16 supplies
// A[m][2]=x, A[m][3]=0. B = ones(4x16) -> D[m][n] = x_m + x_{m+16}.
// D layout: vgpr j = row j (lanes 0-15) / row j+8 (lanes 16-31), so summing
// the 8 D vgprs gives Sum(rows 0-7) in lanes 0-15 and Sum(rows 8-15) in
// lanes 16-31; one shfl_xor(16) completes the 32-lane total (in all lanes).
// Requires EXEC == all 1s: call only from fully-converged code.
__device__ __forceinline__ float wave_allreduce_wmma(float x) {
    v2f a; a[0] = x;    a[1] = 0.0f;
    v2f b; b[0] = 1.0f; b[1] = 1.0f;
    v8f c = {};
    c = __builtin_amdgcn_wmma_f32_16x16x4_f32(
        /*neg_a=*/false, a, /*neg_b=*/false, b,
        /*c_mod=*/(short)0, c, /*reuse_a=*/false, /*reuse_b=*/false);
    float s = c[0] + c[1] + c[2] + c[3] + c[4] + c[5] + c[6] + c[7];
    s += __shfl_xor(s, 16, 32);
    return s;
}

// ws layout: [0,40)   : 5 f64 accumulators {sum_p0, sumsq_p0, sum_p1, sumsq_p1, kcl_sumsq}
//            [64, 64+4*N_NODES) : f32 node_sum
__global__ void pl_init_kernel(float* wsf) {
    int i = blockIdx.x * blockDim.x + threadIdx.x;
    if (i < 16 + N_NODES) wsf[i] = 0.0f;   // first 16 floats cover the f64 accs + pad
}

__global__ void pl_edge_kernel(const float* __restrict__ nf,
                               const int*   __restrict__ ei,
                               const float* __restrict__ logits,
                               const float* __restrict__ params,
                               float*       __restrict__ node_sum,
                               double*      __restrict__ acc) {
    int tid    = blockIdx.x * blockDim.x + threadIdx.x;
    int stride = gridDim.x * blockDim.x;

    float s0 = 0.0f, q0 = 0.0f, s1 = 0.0f, q1 = 0.0f;

    const float2* p2 = (const float2*)params;
    for (int i = tid; i < N_EDGES; i += stride) {
        int s = ei[i];
        int d = ei[N_EDGES + i];
        float  vs = nf[4 * s];          // node_features[:,0], gather (L2-resident)
        float  vd = nf[4 * d];
        float2 p  = p2[i];
        float  prob = 1.0f / (1.0f + expf(-logits[i]));
        float  cur  = fabsf(vs - vd) / (p.x + p.y + EPS_V) * prob;

        unsafeAtomicAdd(node_sum + d,  cur);   // global_atomic_add_f32, L2 scope
        unsafeAtomicAdd(node_sum + s, -cur);

        s0 += p.x; q0 += p.x * p.x;
        s1 += p.y; q1 += p.y * p.y;
    }

    // fully converged here -> EXEC all ones, WMMA legal
    s0 = wave_allreduce_wmma(s0);
    q0 = wave_allreduce_wmma(q0);
    s1 = wave_allreduce_wmma(s1);
    q1 = wave_allreduce_wmma(q1);

    if ((threadIdx.x & 31) == 0) {
        unsafeAtomicAdd(acc + 0, (double)s0);
        unsafeAtomicAdd(acc + 1, (double)q0);
        unsafeAtomicAdd(acc + 2, (double)s1);
        unsafeAtomicAdd(acc + 3, (double)q1);
    }
}

__global__ void pl_node_kernel(const float* __restrict__ node_sum,
                               double*      __restrict__ acc) {
    int tid    = blockIdx.x * blockDim.x + threadIdx.x;
    int stride = gridDim.x * blockDim.x;

    float sq = 0.0f;
    for (int i = tid; i < N_NODES; i += stride) {
        float ns = node_sum[i];
        sq += ns * ns;
    }
    sq = wave_allreduce_wmma(sq);
    if ((threadIdx.x & 31) == 0)
        unsafeAtomicAdd(acc + 4, (double)sq);
}

__global__ void pl_finalize_kernel(const double* __restrict__ acc,
                                   float*        __restrict__ out) {
    if (blockIdx.x == 0 && threadIdx.x == 0) {
        const double E = (double)N_EDGES;
        const double N = (double)N_NODES;
        double var0 = (acc[1] - acc[0] * acc[0] / E) / (E - 1.0);
        double var1 = (acc[3] - acc[2] * acc[2] / E) / (E - 1.0);
        double kvl  = 0.5 * (var0 + var1);
        double kcl  = acc[4] / N;
        out[0] = (float)(kcl + kvl);   // ALPHA_KCL = ALPHA_KVL = 1
    }
}

extern "C" void kernel_launch(void* const* d_in, const int* in_sizes, int n_in,
                              void* d_out, int out_size, void* d_ws, size_t ws_size,
                              hipStream_t stream) {
    const float* nf     = (const float*)d_in[0];  // (N_NODES, 4) f32
    const int*   ei     = (const int*)  d_in[1];  // (2, N_EDGES) i32
    const float* logits = (const float*)d_in[2];  // (N_EDGES,)   f32
    const float* params = (const float*)d_in[3];  // (N_EDGES, 2) f32
    float*       out    = (float*)d_out;

    double* acc      = (double*)d_ws;
    float*  node_sum = (float*)((char*)d_ws + 64);

    pl_init_kernel<<<(N_NODES + 16 + 255) / 256, 256, 0, stream>>>((float*)d_ws);
    pl_edge_kernel<<<1024, 256, 0, stream>>>(nf, ei, logits, params, node_sum, acc);
    pl_node_kernel<<<256, 256, 0, stream>>>(node_sum, acc);
    pl_finalize_kernel<<<1, 32, 0, stream>>>(acc, out);
}